// IrradianceVolumes_16149077033154
// MI455X (gfx1250) — compile-verified
//
#include <hip/hip_runtime.h>

typedef float v2f __attribute__((ext_vector_type(2)));
typedef float v8f __attribute__((ext_vector_type(8)));

#define RES 128

// SH basis as a linear map from 20 monomials (deg<=3 in normal x,y,z) to the
// 16 degree-4 real-SH components:  sh[m] = sum_k mono[k] * SHT[k][m]
// Monomial order:
//   0:1 1:x 2:y 3:z 4:xx 5:yy 6:zz 7:xy 8:yz 9:xz
//  10:x3 11:x2y 12:x2z 13:xy2 14:xyz 15:xz2 16:y3 17:y2z 18:yz2 19:z3
__constant__ float c_SHT[20 * 16] = {
/*1  */ 0.28209479177387814f,0,0,0, 0,0,0,0, 0,0,0,0, 0,0,0,0,
/*x  */ 0,0,0,-0.4886025119029199f, 0,0,0,0, 0,0,0,0, 0,0,0,0,
/*y  */ 0,-0.4886025119029199f,0,0, 0,0,0,0, 0,0,0,0, 0,0,0,0,
/*z  */ 0,0,0.4886025119029199f,0, 0,0,0,0, 0,0,0,0, 0,0,0,0,
/*xx */ 0,0,0,0, 0,0,-0.31539156525252005f,0, 0.5462742152960396f,0,0,0, 0,0,0,0,
/*yy */ 0,0,0,0, 0,0,-0.31539156525252005f,0, -0.5462742152960396f,0,0,0, 0,0,0,0,
/*zz */ 0,0,0,0, 0,0,0.6307831305050401f,0, 0,0,0,0, 0,0,0,0,
/*xy */ 0,0,0,0, 1.0925484305920792f,0,0,0, 0,0,0,0, 0,0,0,0,
/*yz */ 0,0,0,0, 0,-1.0925484305920792f,0,0, 0,0,0,0, 0,0,0,0,
/*xz */ 0,0,0,0, 0,0,0,-1.0925484305920792f, 0,0,0,0, 0,0,0,0,
/*x3 */ 0,0,0,0, 0,0,0,0, 0,0,0,0, 0,0.4570457994644658f,0,-0.5900435899266435f,
/*x2y*/ 0,0,0,0, 0,0,0,0, 0,-1.7701307697799304f,0,0.4570457994644658f, 0,0,0,0,
/*x2z*/ 0,0,0,0, 0,0,0,0, 0,0,0,0, -1.1195289977703462f,0,1.445305721320277f,0,
/*xy2*/ 0,0,0,0, 0,0,0,0, 0,0,0,0, 0,0.4570457994644658f,0,1.7701307697799304f,
/*xyz*/ 0,0,0,0, 0,0,0,0, 0,0,2.890611442640554f,0, 0,0,0,0,
/*xz2*/ 0,0,0,0, 0,0,0,0, 0,0,0,0, 0,-1.8281831978578632f,0,0,
/*y3 */ 0,0,0,0, 0,0,0,0, 0,0.5900435899266435f,0,0.4570457994644658f, 0,0,0,0,
/*y2z*/ 0,0,0,0, 0,0,0,0, 0,0,0,0, -1.1195289977703462f,0,-1.445305721320277f,0,
/*yz2*/ 0,0,0,0, 0,0,0,0, 0,0,0,-1.8281831978578632f, 0,0,0,0,
/*z3 */ 0,0,0,0, 0,0,0,0, 0,0,0,0, 0.7463526651802308f,0,0,0
};

__global__ __launch_bounds__(256)
void IrradianceVolumes_kernel(const float* __restrict__ coeff,
                              const float* __restrict__ aabb,
                              const float* __restrict__ pts,
                              const float* __restrict__ nrms,
                              float* __restrict__ out, int npts)
{
    const int lane = threadIdx.x & 31;
    const int hi   = lane >> 4;      // 0: SH comps 0-7, 1: comps 8-15
    const int col  = lane & 15;      // point slot within wave == matrix column
    const int wid  = blockIdx.x * (blockDim.x >> 5) + (threadIdx.x >> 5);

    int pt = wid * 16 + col;
    const bool valid = (pt < npts);
    if (!valid) pt = npts - 1;       // clamp: keep EXEC all-ones for WMMA

    // ---- point & normal -------------------------------------------------
    const float px = pts[pt*3+0], py = pts[pt*3+1], pz = pts[pt*3+2];
    const float nx = nrms[pt*3+0], ny = nrms[pt*3+1], nz = nrms[pt*3+2];

    const float a0 = aabb[0], a1 = aabb[1], a2 = aabb[2];
    const float a3 = aabb[3], a4 = aabb[4], a5 = aabb[5];

    float gx = (px - a0) / (a3 - a0) * (float)(RES - 1);
    float gy = (py - a1) / (a4 - a1) * (float)(RES - 1);
    float gz = (pz - a2) / (a5 - a2) * (float)(RES - 1);
    gx = fminf(fmaxf(gx, 0.f), (float)(RES - 1));
    gy = fminf(fmaxf(gy, 0.f), (float)(RES - 1));
    gz = fminf(fmaxf(gz, 0.f), (float)(RES - 1));
    int ix = (int)floorf(gx); ix = ix < 0 ? 0 : (ix > RES-2 ? RES-2 : ix);
    int iy = (int)floorf(gy); iy = iy < 0 ? 0 : (iy > RES-2 ? RES-2 : iy);
    int iz = (int)floorf(gz); iz = iz < 0 ? 0 : (iz > RES-2 ? RES-2 : iz);
    const float fx = gx - (float)ix, fy = gy - (float)iy, fz = gz - (float)iz;

    // ---- SH basis via WMMA: sh = SHT^T (16x20) * mono (20 x 16 points) --
    const float xx = nx*nx, yy = ny*ny, zz = nz*nz;
    const float xy = nx*ny, yz = ny*nz, xz = nx*nz;

    // B chunks (K=4 each): low lanes hold K=4t,4t+1; high lanes K=4t+2,4t+3
    const v2f B0 = { hi ? ny    : 1.0f , hi ? nz    : nx    };  // 1  x  | y  z
    const v2f B1 = { hi ? zz    : xx   , hi ? xy    : yy    };  // xx yy | zz xy
    const v2f B2 = { hi ? xx*nx : yz   , hi ? xx*ny : xz    };  // yz xz | x3 x2y
    const v2f B3 = { hi ? xy*nz : xx*nz, hi ? nx*zz : nx*yy };  // x2z xy2 | xyz xz2
    const v2f B4 = { hi ? ny*zz : yy*ny, hi ? zz*nz : yy*nz };  // y3 y2z | yz2 z3

    // A chunks: constant SH matrix, row m = SH component (= lane&15)
    const int kb = 2 * hi;
    const v2f A0 = { c_SHT[(0  + kb)*16 + col], c_SHT[(1  + kb)*16 + col] };
    const v2f A1 = { c_SHT[(4  + kb)*16 + col], c_SHT[(5  + kb)*16 + col] };
    const v2f A2 = { c_SHT[(8  + kb)*16 + col], c_SHT[(9  + kb)*16 + col] };
    const v2f A3 = { c_SHT[(12 + kb)*16 + col], c_SHT[(13 + kb)*16 + col] };
    const v2f A4 = { c_SHT[(16 + kb)*16 + col], c_SHT[(17 + kb)*16 + col] };

    v8f sh = {0.f,0.f,0.f,0.f,0.f,0.f,0.f,0.f};
    sh = __builtin_amdgcn_wmma_f32_16x16x4_f32(false, A0, false, B0, (short)0, sh, false, false);
    sh = __builtin_amdgcn_wmma_f32_16x16x4_f32(false, A1, false, B1, (short)0, sh, false, false);
    sh = __builtin_amdgcn_wmma_f32_16x16x4_f32(false, A2, false, B2, (short)0, sh, false, false);
    sh = __builtin_amdgcn_wmma_f32_16x16x4_f32(false, A3, false, B3, (short)0, sh, false, false);
    sh = __builtin_amdgcn_wmma_f32_16x16x4_f32(false, A4, false, B4, (short)0, sh, false, false);
    // D layout: VGPR r, lane L -> sh[comp 8*hi + r] of point (L&15)  ✓ matches gather below

    // ---- gather 8 corners, fold trilerp weight + SH dot per lane --------
    float o0 = 0.f, o1 = 0.f, o2 = 0.f;
    #pragma unroll
    for (int dx = 0; dx < 2; ++dx) {
        const float wxv = dx ? fx : 1.f - fx;
        #pragma unroll
        for (int dy = 0; dy < 2; ++dy) {
            const float wxy = wxv * (dy ? fy : 1.f - fy);
            const size_t rowcell = ((size_t)(ix + dx) * RES + (size_t)(iy + dy)) * RES + (size_t)iz;
            const float* base = coeff + rowcell * 48 + hi * 24;  // comps 8*hi..8*hi+7, 3 ch
            #pragma unroll
            for (int dz = 0; dz < 2; ++dz) {
                const float w = wxy * (dz ? fz : 1.f - fz);
                const float4* cp = (const float4*)(base + dz * 48);
                float4 q0 = cp[0], q1 = cp[1], q2 = cp[2], q3 = cp[3], q4 = cp[4], q5 = cp[5];
                float v[24];
                *(float4*)&v[0]  = q0; *(float4*)&v[4]  = q1; *(float4*)&v[8]  = q2;
                *(float4*)&v[12] = q3; *(float4*)&v[16] = q4; *(float4*)&v[20] = q5;
                float t0 = 0.f, t1 = 0.f, t2 = 0.f;
                #pragma unroll
                for (int r = 0; r < 8; ++r) {
                    t0 += sh[r] * v[r*3+0];
                    t1 += sh[r] * v[r*3+1];
                    t2 += sh[r] * v[r*3+2];
                }
                o0 += w * t0; o1 += w * t1; o2 += w * t2;
            }
        }
    }

    // ---- merge the two comp-halves (lane L <-> lane L+16), ReLU, store --
    o0 += __shfl_xor(o0, 16, 32);
    o1 += __shfl_xor(o1, 16, 32);
    o2 += __shfl_xor(o2, 16, 32);

    if (valid && hi == 0) {
        out[(size_t)pt*3+0] = fmaxf(o0, 0.f);
        out[(size_t)pt*3+1] = fmaxf(o1, 0.f);
        out[(size_t)pt*3+2] = fmaxf(o2, 0.f);
    }
}

extern "C" void kernel_launch(void* const* d_in, const int* in_sizes, int n_in,
                              void* d_out, int out_size, void* d_ws, size_t ws_size,
                              hipStream_t stream) {
    const float* coeff = (const float*)d_in[0];
    const float* aabb  = (const float*)d_in[1];
    const float* pts   = (const float*)d_in[2];
    const float* nrms  = (const float*)d_in[3];
    float* out = (float*)d_out;

    const int npts = in_sizes[2] / 3;
    const int ptsPerBlock = 128;                 // 8 waves * 16 points/wave
    const int grid = (npts + ptsPerBlock - 1) / ptsPerBlock;

    IrradianceVolumes_kernel<<<grid, 256, 0, stream>>>(coeff, aabb, pts, nrms, out, npts);
}